// Transformer_395136991551
// MI455X (gfx1250) — compile-verified
//
#include <hip/hip_runtime.h>
#include <cstdint>

typedef __attribute__((ext_vector_type(2))) float v2f;
typedef __attribute__((ext_vector_type(8))) float v8f;
typedef int vsi4 __attribute__((vector_size(16)));   // matches builtin param pointee

#define TB 8
#define TD 512
#define TN 1024
#define TH 4
#define TDK 128
#define TDFF 1024
#define TM (TB * TN)          // 8192 rows
#define LN_EPS 1e-6f
#define SCORE_SCALE 0.08838834764831843f  // 1/sqrt(128)

#define PADA 36   // LDS row stride for A tile (64 x 32)
#define PADB 68   // LDS row stride for B tile (32 x 64)

#if defined(__AMDGCN__) && __has_builtin(__builtin_amdgcn_global_load_async_to_lds_b128)
#define USE_ASYNC 1
#else
#define USE_ASYNC 0
#endif

#if USE_ASYNC
__device__ __forceinline__ void async_cp16(const float* g, float* l)
{
    // 16B global -> LDS async copy (per-lane addresses), tracked by ASYNCcnt.
    __builtin_amdgcn_global_load_async_to_lds_b128(
        (vsi4*)g, (vsi4*)l, 0, 0);
}
__device__ __forceinline__ void wait_async_le8()
{
#if __has_builtin(__builtin_amdgcn_s_wait_asynccnt)
    __builtin_amdgcn_s_wait_asynccnt(8);
#else
    asm volatile("s_wait_asynccnt 0x8" ::: "memory");
#endif
}
__device__ __forceinline__ void wait_async_0()
{
#if __has_builtin(__builtin_amdgcn_s_wait_asynccnt)
    __builtin_amdgcn_s_wait_asynccnt(0);
#else
    asm volatile("s_wait_asynccnt 0x0" ::: "memory");
#endif
}
#endif

// ---------------------------------------------------------------------------
// Generic fp32 WMMA GEMM core. Block = 128 threads (4 waves).
// Block tile 64(M) x 64(N), K-chunk 32, each wave computes 16x64 (4 accums).
// Staging: async global->LDS b128 with double buffering when available,
// otherwise float4 register staging with fetch-ahead pipelining.
// All dims are multiples of 64 in this model (no edge handling needed).
// ---------------------------------------------------------------------------
__device__ __forceinline__ void gemm_core(
    const float* __restrict__ A, int lda,
    const float* __restrict__ Bm, int ldb, bool transB,
    float* __restrict__ C, int ldc,
    const float* __restrict__ bias,
    const float* __restrict__ resid, int ldr,
    int K, float alpha, bool relu)
{
    const int tN = blockIdx.x * 64;
    const int tM = blockIdx.y * 64;

    const int tid  = threadIdx.x;          // 0..127
    const int wave = tid >> 5;             // 0..3
    const int lane = tid & 31;
    const int hi   = lane >> 4;            // 0/1
    const int lm   = lane & 15;

#if USE_ASYNC
    __shared__ float As[2][64 * PADA];
    __shared__ float Bs[2][32 * PADB];
#else
    __shared__ float As[1][64 * PADA];
    __shared__ float Bs[1][32 * PADB];
#endif

    v8f acc0 = {0,0,0,0,0,0,0,0};
    v8f acc1 = {0,0,0,0,0,0,0,0};
    v8f acc2 = {0,0,0,0,0,0,0,0};
    v8f acc3 = {0,0,0,0,0,0,0,0};

    const int nT = K >> 5;   // number of 32-wide K chunks

    // WMMA compute over one staged 64x32 (A) / 32x64 (B) LDS tile.
    auto compute = [&](const float* Asb, const float* Bsb) {
        const float* arow = &Asb[(wave * 16 + lm) * PADA];
#pragma unroll
        for (int kk = 0; kk < 32; kk += 4) {
            v2f a;
            a.x = arow[kk + 2 * hi];
            a.y = arow[kk + 2 * hi + 1];
            const float* b0 = &Bsb[(kk + 2 * hi) * PADB + lm];
            const float* b1 = &Bsb[(kk + 2 * hi + 1) * PADB + lm];
            v2f bf;
            bf.x = b0[0];  bf.y = b1[0];
            acc0 = __builtin_amdgcn_wmma_f32_16x16x4_f32(false, a, false, bf, (short)0, acc0, false, false);
            bf.x = b0[16]; bf.y = b1[16];
            acc1 = __builtin_amdgcn_wmma_f32_16x16x4_f32(false, a, false, bf, (short)0, acc1, false, false);
            bf.x = b0[32]; bf.y = b1[32];
            acc2 = __builtin_amdgcn_wmma_f32_16x16x4_f32(false, a, false, bf, (short)0, acc2, false, false);
            bf.x = b0[48]; bf.y = b1[48];
            acc3 = __builtin_amdgcn_wmma_f32_16x16x4_f32(false, a, false, bf, (short)0, acc3, false, false);
        }
    };

#if USE_ASYNC
    if (!transB) {
        // Async double-buffered pipeline: 8 x b128 async copies per wave/tile.
        auto issue = [&](int t, int buf) {
            const long long k0 = (long long)t << 5;
#pragma unroll
            for (int p = 0; p < 4; ++p) {            // A tile: 64 x 32
                int s = tid + p * 128;
                int r = s >> 3, q = s & 7;
                async_cp16(A + (long long)(tM + r) * lda + k0 + q * 4,
                           &As[buf][r * PADA + q * 4]);
            }
#pragma unroll
            for (int p = 0; p < 4; ++p) {            // B tile: 32 x 64
                int s = tid + p * 128;
                int kk = s >> 4, c4 = s & 15;
                async_cp16(Bm + (k0 + kk) * (long long)ldb + tN + c4 * 4,
                           &Bs[buf][kk * PADB + c4 * 4]);
            }
        };
        issue(0, 0);
        for (int t = 0; t < nT; ++t) {
            const int buf = t & 1;
            if (t + 1 < nT) {
                issue(t + 1, buf ^ 1);   // overlap next tile with compute
                wait_async_le8();        // tile t (8 older ops) complete
            } else {
                wait_async_0();
            }
            __syncthreads();
            compute(As[buf], Bs[buf]);
            __syncthreads();
        }
    } else
#endif
    {
        // Register-staged pipeline: float4 global loads, b128 LDS stores.
        float4 ra[4], rb[4];
        auto fetch = [&](int t) {
            const long long k0 = (long long)t << 5;
#pragma unroll
            for (int p = 0; p < 4; ++p) {
                int s = tid + p * 128;
                int r = s >> 3, q = s & 7;
                ra[p] = *(const float4*)(A + (long long)(tM + r) * lda + k0 + q * 4);
            }
            if (!transB) {
#pragma unroll
                for (int p = 0; p < 4; ++p) {
                    int s = tid + p * 128;
                    int kk = s >> 4, c4 = s & 15;
                    rb[p] = *(const float4*)(Bm + (k0 + kk) * (long long)ldb + tN + c4 * 4);
                }
            } else {
#pragma unroll
                for (int p = 0; p < 4; ++p) {
                    int s = tid + p * 128;
                    int c = s >> 3, q = s & 7;
                    rb[p] = *(const float4*)(Bm + (long long)(tN + c) * ldb + k0 + q * 4);
                }
            }
        };
        auto stage = [&]() {
#pragma unroll
            for (int p = 0; p < 4; ++p) {
                int s = tid + p * 128;
                int r = s >> 3, q = s & 7;
                *(float4*)&As[0][r * PADA + q * 4] = ra[p];
            }
            if (!transB) {
#pragma unroll
                for (int p = 0; p < 4; ++p) {
                    int s = tid + p * 128;
                    int kk = s >> 4, c4 = s & 15;
                    *(float4*)&Bs[0][kk * PADB + c4 * 4] = rb[p];
                }
            } else {
#pragma unroll
                for (int p = 0; p < 4; ++p) {
                    int s = tid + p * 128;
                    int c = s >> 3, q = s & 7;
                    Bs[0][(q * 4 + 0) * PADB + c] = rb[p].x;
                    Bs[0][(q * 4 + 1) * PADB + c] = rb[p].y;
                    Bs[0][(q * 4 + 2) * PADB + c] = rb[p].z;
                    Bs[0][(q * 4 + 3) * PADB + c] = rb[p].w;
                }
            }
        };
        fetch(0);
        for (int t = 0; t < nT; ++t) {
            __syncthreads();
            stage();
            __syncthreads();
            if (t + 1 < nT) fetch(t + 1);   // overlap next loads with compute
            compute(&As[0][0], &Bs[0][0]);
        }
    }

    // Epilogue. C/D layout: VGPR r -> row = r + 8*hi (within 16), col = lm.
#pragma unroll
    for (int t = 0; t < 4; ++t) {
        v8f acc = (t == 0) ? acc0 : (t == 1) ? acc1 : (t == 2) ? acc2 : acc3;
        int col = tN + t * 16 + lm;
#pragma unroll
        for (int r = 0; r < 8; ++r) {
            int row = tM + wave * 16 + hi * 8 + r;
            float v = acc[r] * alpha;
            if (bias)  v += bias[col];
            if (resid) v += resid[(long long)row * ldr + col];
            if (relu)  v = fmaxf(v, 0.0f);
            C[(long long)row * ldc + col] = v;
        }
    }
}

// Plain GEMM: C[M,Nc] = A[M,K] @ W[K,Nc] + bias (+resid) (relu)
__global__ __launch_bounds__(128) void k_gemm(
    const float* __restrict__ A, const float* __restrict__ W,
    float* __restrict__ C, const float* __restrict__ bias,
    const float* __restrict__ resid, int K, int Nc, int relu)
{
    gemm_core(A, K, W, Nc, false, C, Nc, bias, resid, Nc, K, 1.0f, relu != 0);
}

// Attention scores: S[z,q,k] = (Q_z @ K_z^T) / sqrt(dk), z = b*H + h
__global__ __launch_bounds__(128) void k_scores(
    const float* __restrict__ Q, const float* __restrict__ Kmat,
    float* __restrict__ S)
{
    int z = blockIdx.z;
    int b = z / TH, h = z % TH;
    const float* Aq = Q    + (long long)b * TN * TD + h * TDK;
    const float* Bk = Kmat + (long long)b * TN * TD + h * TDK;
    float* Sz = S + (long long)z * TN * TN;
    gemm_core(Aq, TD, Bk, TD, true, Sz, TN, nullptr, nullptr, 0,
              TDK, SCORE_SCALE, false);
}

// O[b, q, h*dk + c] = sum_k P[z,q,k] * V[b,k,h*dk+c]
__global__ __launch_bounds__(128) void k_pv(
    const float* __restrict__ S, const float* __restrict__ V,
    float* __restrict__ O)
{
    int z = blockIdx.z;
    int b = z / TH, h = z % TH;
    const float* Ap = S + (long long)z * TN * TN;
    const float* Bv = V + (long long)b * TN * TD + h * TDK;
    float* Co = O + (long long)b * TN * TD + h * TDK;
    gemm_core(Ap, TN, Bv, TD, false, Co, TD, nullptr, nullptr, 0,
              TN, 1.0f, false);
}

// ---------------------------------------------------------------------------
// LayerNorm (torch-style): y = g*(x-mean)/(std+eps)+b, std unbiased (ddof=1)
// One block (256 thr) per row of D=512.
// ---------------------------------------------------------------------------
__global__ __launch_bounds__(256) void k_ln(
    const float* __restrict__ x, const float* __restrict__ g,
    const float* __restrict__ bvec, float* __restrict__ y)
{
    __shared__ float s_sum[256];
    __shared__ float s_sq[256];
    const long long row = blockIdx.x;
    const int tid = threadIdx.x;
    float v0 = x[row * TD + tid];
    float v1 = x[row * TD + tid + 256];
    s_sum[tid] = v0 + v1;
    s_sq[tid]  = v0 * v0 + v1 * v1;
    __syncthreads();
#pragma unroll
    for (int off = 128; off > 0; off >>= 1) {
        if (tid < off) {
            s_sum[tid] += s_sum[tid + off];
            s_sq[tid]  += s_sq[tid + off];
        }
        __syncthreads();
    }
    float mean = s_sum[0] * (1.0f / TD);
    float var  = (s_sq[0] - (float)TD * mean * mean) * (1.0f / (TD - 1));
    float inv  = 1.0f / (sqrtf(fmaxf(var, 0.0f)) + LN_EPS);
    y[row * TD + tid]       = g[tid]       * (v0 - mean) * inv + bvec[tid];
    y[row * TD + tid + 256] = g[tid + 256] * (v1 - mean) * inv + bvec[tid + 256];
}

// ---------------------------------------------------------------------------
// Row softmax over N=1024 columns. One block (256 thr) per row.
// ---------------------------------------------------------------------------
__global__ __launch_bounds__(256) void k_softmax(float* __restrict__ S)
{
    __shared__ float red[256];
    const long long row = blockIdx.x;
    float* p = S + row * TN;
    const int tid = threadIdx.x;
    float v[4];
#pragma unroll
    for (int j = 0; j < 4; ++j) v[j] = p[tid + j * 256];
    float m = fmaxf(fmaxf(v[0], v[1]), fmaxf(v[2], v[3]));
    red[tid] = m;
    __syncthreads();
#pragma unroll
    for (int off = 128; off > 0; off >>= 1) {
        if (tid < off) red[tid] = fmaxf(red[tid], red[tid + off]);
        __syncthreads();
    }
    m = red[0];
    __syncthreads();
    float s = 0.0f;
#pragma unroll
    for (int j = 0; j < 4; ++j) { v[j] = __expf(v[j] - m); s += v[j]; }
    red[tid] = s;
    __syncthreads();
#pragma unroll
    for (int off = 128; off > 0; off >>= 1) {
        if (tid < off) red[tid] += red[tid + off];
        __syncthreads();
    }
    float invs = 1.0f / red[0];
#pragma unroll
    for (int j = 0; j < 4; ++j) p[tid + j * 256] = v[j] * invs;
}

// ---------------------------------------------------------------------------
// Transposes: [B, D, N] <-> row-major [B*N, D]
// ---------------------------------------------------------------------------
__global__ __launch_bounds__(256) void k_transpose_in(
    const float* __restrict__ src, float* __restrict__ x)
{
    __shared__ float tile[32][33];
    int n0 = blockIdx.x * 32, d0 = blockIdx.y * 32, b = blockIdx.z;
    int tx = threadIdx.x, ty = threadIdx.y;     // (32, 8)
#pragma unroll
    for (int i = 0; i < 4; ++i)
        tile[ty + i * 8][tx] = src[((long long)b * TD + d0 + ty + i * 8) * TN + n0 + tx];
    __syncthreads();
#pragma unroll
    for (int i = 0; i < 4; ++i)
        x[((long long)b * TN + n0 + ty + i * 8) * TD + d0 + tx] = tile[tx][ty + i * 8];
}

__global__ __launch_bounds__(256) void k_transpose_out(
    const float* __restrict__ y, float* __restrict__ out)
{
    __shared__ float tile[32][33];
    int n0 = blockIdx.x * 32, d0 = blockIdx.y * 32, b = blockIdx.z;
    int tx = threadIdx.x, ty = threadIdx.y;
#pragma unroll
    for (int i = 0; i < 4; ++i)
        tile[ty + i * 8][tx] = y[((long long)b * TN + n0 + ty + i * 8) * TD + d0 + tx];
    __syncthreads();
#pragma unroll
    for (int i = 0; i < 4; ++i)
        out[((long long)b * TD + d0 + ty + i * 8) * TN + n0 + tx] = tile[tx][ty + i * 8];
}

// ---------------------------------------------------------------------------
// Host-side orchestration
// ---------------------------------------------------------------------------
namespace {

struct Work {
    float *A_enc, *A_x, *A_mem, *A_ln, *A_q, *A_k, *A_v, *A_t2, *A_t1, *S;
};

inline void launch_gemm(const float* A, const float* W, float* C,
                        const float* bias, const float* resid,
                        int K, int Nc, int relu, hipStream_t st)
{
    dim3 grid(Nc / 64, TM / 64, 1);
    k_gemm<<<grid, 128, 0, st>>>(A, W, C, bias, resid, K, Nc, relu);
}

inline void launch_ln(const float* x, const float* g, const float* b,
                      float* y, hipStream_t st)
{
    k_ln<<<TM, 256, 0, st>>>(x, g, b, y);
}

// qin: [M,D] query source; kvin: [M,D] key/value source; x: residual in/out
inline void mha(const Work& w, const float* qin, const float* kvin,
                const float* W, const float* bv, float* x, hipStream_t st)
{
    launch_gemm(qin,  W + 0 * TD * TD, w.A_q, bv + 0 * TD, nullptr, TD, TD, 0, st);
    launch_gemm(kvin, W + 1 * TD * TD, w.A_k, bv + 1 * TD, nullptr, TD, TD, 0, st);
    launch_gemm(kvin, W + 2 * TD * TD, w.A_v, bv + 2 * TD, nullptr, TD, TD, 0, st);
    dim3 gs(TN / 64, TN / 64, TB * TH);
    k_scores<<<gs, 128, 0, st>>>(w.A_q, w.A_k, w.S);
    k_softmax<<<TB * TH * TN, 256, 0, st>>>(w.S);
    dim3 gp(TDK / 64, TN / 64, TB * TH);
    k_pv<<<gp, 128, 0, st>>>(w.S, w.A_v, w.A_t2);
    launch_gemm(w.A_t2, W + 3 * TD * TD, x, bv + 3 * TD, x, TD, TD, 0, st);
}

inline void run_model(
    const Work& w,
    const float* src, const float* tgt, float* out,
    const float* eaw, const float* eab,
    const float* efw1, const float* efb1, const float* efw2, const float* efb2,
    const float* elg, const float* elb,
    const float* dsw, const float* dsb, const float* dcw, const float* dcb,
    const float* dfw1, const float* dfb1, const float* dfw2, const float* dfb2,
    const float* dlg, const float* dlb,
    hipStream_t st)
{
    dim3 tgrid(TN / 32, TD / 32, TB), tblk(32, 8);
    k_transpose_in<<<tgrid, tblk, 0, st>>>(src, w.A_enc);
    k_transpose_in<<<tgrid, tblk, 0, st>>>(tgt, w.A_x);

    // ---- Encoder ----
    launch_ln(w.A_enc, elg + 0 * TD, elb + 0 * TD, w.A_ln, st);
    mha(w, w.A_ln, w.A_ln, eaw, eab, w.A_enc, st);
    launch_ln(w.A_enc, elg + 1 * TD, elb + 1 * TD, w.A_ln, st);
    launch_gemm(w.A_ln, efw1, w.A_t1, efb1, nullptr, TD, TDFF, 1, st);
    launch_gemm(w.A_t1, efw2, w.A_enc, efb2, w.A_enc, TDFF, TD, 0, st);
    launch_ln(w.A_enc, elg + 2 * TD, elb + 2 * TD, w.A_mem, st);

    // ---- Decoder ----
    launch_ln(w.A_x, dlg + 0 * TD, dlb + 0 * TD, w.A_ln, st);
    mha(w, w.A_ln, w.A_ln, dsw, dsb, w.A_x, st);
    launch_ln(w.A_x, dlg + 1 * TD, dlb + 1 * TD, w.A_ln, st);
    mha(w, w.A_ln, w.A_mem, dcw, dcb, w.A_x, st);
    launch_ln(w.A_x, dlg + 2 * TD, dlb + 2 * TD, w.A_ln, st);
    launch_gemm(w.A_ln, dfw1, w.A_t1, dfb1, nullptr, TD, TDFF, 1, st);
    launch_gemm(w.A_t1, dfw2, w.A_x, dfb2, w.A_x, TDFF, TD, 0, st);
    launch_ln(w.A_x, dlg + 3 * TD, dlb + 3 * TD, w.A_ln, st);

    k_transpose_out<<<tgrid, tblk, 0, st>>>(w.A_ln, out);
}

} // namespace

extern "C" void kernel_launch(void* const* d_in, const int* in_sizes, int n_in,
                              void* d_out, int out_size, void* d_ws, size_t ws_size,
                              hipStream_t stream)
{
    (void)in_sizes; (void)n_in; (void)out_size; (void)ws_size;

    const float* src  = (const float*)d_in[0];
    const float* tgt  = (const float*)d_in[1];
    const float* eaw  = (const float*)d_in[2];
    const float* eab  = (const float*)d_in[3];
    const float* efw1 = (const float*)d_in[4];
    const float* efb1 = (const float*)d_in[5];
    const float* efw2 = (const float*)d_in[6];
    const float* efb2 = (const float*)d_in[7];
    const float* elg  = (const float*)d_in[8];
    const float* elb  = (const float*)d_in[9];
    const float* dsw  = (const float*)d_in[10];
    const float* dsb  = (const float*)d_in[11];
    const float* dcw  = (const float*)d_in[12];
    const float* dcb  = (const float*)d_in[13];
    const float* dfw1 = (const float*)d_in[14];
    const float* dfb1 = (const float*)d_in[15];
    const float* dfw2 = (const float*)d_in[16];
    const float* dfb2 = (const float*)d_in[17];
    const float* dlg  = (const float*)d_in[18];
    const float* dlb  = (const float*)d_in[19];

    float* out = (float*)d_out;
    float* ws  = (float*)d_ws;

    const size_t ME = (size_t)TM * TD;       // 4,194,304
    const size_t MF = (size_t)TM * TDFF;     // 8,388,608
    Work w;
    size_t off = 0;
    w.A_enc = ws + off; off += ME;
    w.A_x   = ws + off; off += ME;
    w.A_mem = ws + off; off += ME;
    w.A_ln  = ws + off; off += ME;
    w.A_q   = ws + off; off += ME;
    w.A_k   = ws + off; off += ME;
    w.A_v   = ws + off; off += ME;
    w.A_t2  = ws + off; off += ME;
    w.A_t1  = ws + off; off += MF;
    w.S     = ws + off; off += (size_t)TB * TH * TN * TN;

    const size_t BDN = (size_t)TB * TD * TN;

    // tgt_embedding = model(src, tgt)  -> second half of d_out
    run_model(w, src, tgt, out + BDN,
              eaw, eab, efw1, efb1, efw2, efb2, elg, elb,
              dsw, dsb, dcw, dcb, dfw1, dfb1, dfw2, dfb2, dlg, dlb, stream);

    // src_embedding = model(tgt, src)  -> first half of d_out
    run_model(w, tgt, src, out,
              eaw, eab, efw1, efb1, efw2, efb2, elg, elb,
              dsw, dsb, dcw, dcb, dfw1, dfb1, dfw2, dfb2, dlg, dlb, stream);
}